// GA_block_15805479650014
// MI455X (gfx1250) — compile-verified
//
#include <hip/hip_runtime.h>
#include <hip/hip_bf16.h>
#include <math.h>

// Problem constants (from reference)
#define NB   64
#define NHW  1024
#define NC   384
#define NC4  (NC / 4)   // 96
#define XROW 1025       // x has HW+1 rows
#define OROW 257        // out has 1 + 256 rows
#define NPART 8         // row-split factor for streaming reductions
#define ROWS_PER_PART (NHW / NPART)   // 128

typedef __attribute__((ext_vector_type(2))) float v2f;
typedef __attribute__((ext_vector_type(4))) float v4f;
typedef __attribute__((ext_vector_type(8))) float v8f;

// ---------------------------------------------------------------------------
// K1a: partial sums of edge_agg over n.  grid (64, 8), block 96 (v4f lanes)
// part[(b*8+p)*NC + c] = sum over 128 rows
// ---------------------------------------------------------------------------
__global__ void k_ea_partial(const float* __restrict__ eagg, float* __restrict__ part) {
    const int b = blockIdx.x, p = blockIdx.y, t = threadIdx.x;  // t < 96
    const v4f* base = (const v4f*)(eagg + (size_t)b * NHW * NC
                                        + (size_t)p * ROWS_PER_PART * NC);
    v4f acc = {};
#pragma unroll 4
    for (int n = 0; n < ROWS_PER_PART; ++n) acc += base[(size_t)n * NC4 + t];
    *(v4f*)(part + (size_t)(b * NPART + p) * NC + 4 * t) = acc;
}

// ---------------------------------------------------------------------------
// K1b: ea[b,c] = (sum_p part) / 1024
// ---------------------------------------------------------------------------
__global__ void k_ea_combine(const float* __restrict__ part, float* __restrict__ ea) {
    const int b = blockIdx.x, c = threadIdx.x;                  // c < 384
    float s = 0.0f;
#pragma unroll
    for (int p = 0; p < NPART; ++p) s += part[(size_t)(b * NPART + p) * NC + c];
    ea[b * NC + c] = s * (1.0f / (float)NHW);
}

// ---------------------------------------------------------------------------
// K2: ci[b,c] = softmax over b of ea[b,c]   (one block, one thread per c)
// ---------------------------------------------------------------------------
__global__ void k_ci_softmax(const float* __restrict__ ea, float* __restrict__ ci) {
    const int c = threadIdx.x;
    float m = -INFINITY;
    for (int b = 0; b < NB; ++b) m = fmaxf(m, ea[b * NC + c]);
    float s = 0.0f;
    for (int b = 0; b < NB; ++b) s += expf(ea[b * NC + c] - m);
    const float inv = 1.0f / s;
    for (int b = 0; b < NB; ++b) ci[b * NC + c] = expf(ea[b * NC + c] - m) * inv;
}

// ---------------------------------------------------------------------------
// K3: ca[b,n] = dot(nodes[b,n,:], ci[b,:])  — one wave per node row, v4f loads
// block = 256 thr (8 waves), grid = (B, HW/8)
// ---------------------------------------------------------------------------
__global__ void k_channel_agg(const float* __restrict__ x,
                              const float* __restrict__ ci,
                              float* __restrict__ ca) {
    __shared__ v4f s_ci[NC4];
    const int b = blockIdx.x;
    for (int i = threadIdx.x; i < NC; i += 256)
        ((float*)s_ci)[i] = ci[b * NC + i];
    __syncthreads();

    const int wid  = threadIdx.x >> 5;
    const int lane = threadIdx.x & 31;
    const int n    = blockIdx.y * 8 + wid;

    const v4f* row = (const v4f*)(x + (size_t)b * XROW * NC + (size_t)(n + 1) * NC);
    float acc = 0.0f;
#pragma unroll
    for (int i = lane; i < NC4; i += 32) {
        const v4f a = row[i];
        const v4f w = s_ci[i];
        acc += a.x * w.x + a.y * w.y + a.z * w.z + a.w * w.w;
    }
#pragma unroll
    for (int off = 16; off > 0; off >>= 1) acc += __shfl_down(acc, off, 32);
    if (lane == 0) ca[b * NHW + n] = acc;
}

// ---------------------------------------------------------------------------
// K4: ni[b,n] = softmax over b of ca[b,n]   (1024 threads total, one per n)
// ---------------------------------------------------------------------------
__global__ void k_ni_softmax(const float* __restrict__ ca, float* __restrict__ ni) {
    const int n = blockIdx.x * blockDim.x + threadIdx.x;
    float m = -INFINITY;
    for (int b = 0; b < NB; ++b) m = fmaxf(m, ca[b * NHW + n]);
    float s = 0.0f;
    for (int b = 0; b < NB; ++b) s += expf(ca[b * NHW + n] - m);
    const float inv = 1.0f / s;
    for (int b = 0; b < NB; ++b) ni[b * NHW + n] = expf(ca[b * NHW + n] - m) * inv;
}

// ---------------------------------------------------------------------------
// K5a: partial weighted sums: sum_{n in part} ni[b,n] * nodes[b,n,c]
// grid (64, 8), block 96 (v4f lanes)
// ---------------------------------------------------------------------------
__global__ void k_na_partial(const float* __restrict__ x,
                             const float* __restrict__ ni,
                             float* __restrict__ part) {
    const int b = blockIdx.x, p = blockIdx.y, t = threadIdx.x;  // t < 96
    const v4f* nodes = (const v4f*)(x + (size_t)b * XROW * NC + NC);
    const float* nip = ni + b * NHW + p * ROWS_PER_PART;
    v4f acc = {};
#pragma unroll 4
    for (int n = 0; n < ROWS_PER_PART; ++n) {
        const float w = nip[n];                  // uniform across block
        const v4f wv = { w, w, w, w };
        acc += wv * nodes[(size_t)(p * ROWS_PER_PART + n) * NC4 + t];
    }
    *(v4f*)(part + (size_t)(b * NPART + p) * NC + 4 * t) = acc;
}

// ---------------------------------------------------------------------------
// K5b: wts[b,c] = sigmoid( sum_p part )
// ---------------------------------------------------------------------------
__global__ void k_na_combine(const float* __restrict__ part, float* __restrict__ wts) {
    const int b = blockIdx.x, c = threadIdx.x;
    float s = 0.0f;
#pragma unroll
    for (int p = 0; p < NPART; ++p) s += part[(size_t)(b * NPART + p) * NC + c];
    wts[b * NC + c] = 1.0f / (1.0f + expf(-s));
}

// ---------------------------------------------------------------------------
// K6 (WMMA): out[b,0,c] = cls[b,c] + (sum_j cls[b,j]*W[c,j]) * wts[b,c]
// FP32 WMMA 16x16x4, one wave per 16x16 tile; grid (64/16, 384/16).
// ---------------------------------------------------------------------------
__global__ void k_cls_wmma(const float* __restrict__ cls,
                           const float* __restrict__ W,
                           const float* __restrict__ w,
                           float* __restrict__ out) {
    const int m0   = blockIdx.x * 16;
    const int n0   = blockIdx.y * 16;
    const int lane = threadIdx.x;        // 0..31
    const int half = lane >> 4;          // 0: K pair (k,k+1), 1: (k+2,k+3)
    const int lm   = lane & 15;

    const float* arow = cls + (size_t)(m0 + lm) * NC;   // A rows
    const float* brow = W   + (size_t)(n0 + lm) * NC;   // W rows give B columns

    v8f acc = {};
#pragma unroll 4
    for (int k = 0; k < NC; k += 4) {
        const int ka = k + 2 * half;
        v2f a  = *(const v2f*)(arow + ka);
        v2f bm = *(const v2f*)(brow + ka);
        acc = __builtin_amdgcn_wmma_f32_16x16x4_f32(
            false, a, false, bm, (short)0, acc, false, false);
    }

#pragma unroll
    for (int v = 0; v < 8; ++v) {
        const int m = m0 + v + 8 * half;          // batch index
        const int c = n0 + lm;                    // channel
        out[(size_t)m * OROW * NC + c] =
            cls[m * NC + c] + acc[v] * w[m * NC + c];
    }
}

// ---------------------------------------------------------------------------
// K7: per-batch stable ascending bitonic sort of (ca, idx), gather last 256
// rows of nodes into out rows 1..256 (v4f copies).  grid = 64, block = 1024.
// ---------------------------------------------------------------------------
__global__ void k_sort_pool(const float* __restrict__ x,
                            const float* __restrict__ ca,
                            float* __restrict__ out) {
    __shared__ float s_val[NHW];
    __shared__ int   s_idx[NHW];
    const int b = blockIdx.x;
    const int t = threadIdx.x;

    s_val[t] = ca[b * NHW + t];
    s_idx[t] = t;
    __syncthreads();

    for (int k = 2; k <= NHW; k <<= 1) {
        for (int j = k >> 1; j > 0; j >>= 1) {
            const int ixj = t ^ j;
            if (ixj > t) {
                const bool up = ((t & k) == 0);
                const float v1 = s_val[t], v2 = s_val[ixj];
                const int   i1 = s_idx[t], i2 = s_idx[ixj];
                // strict "greater" with index tie-break == stable ascending sort
                const bool gt = (v1 > v2) || (v1 == v2 && i1 > i2);
                if (gt == up) {
                    s_val[t] = v2; s_val[ixj] = v1;
                    s_idx[t] = i2; s_idx[ixj] = i1;
                }
            }
            __syncthreads();
        }
    }

    // ascending sort -> largest 256 are entries [768..1023], kept in order
    const v4f* nodes4 = (const v4f*)(x   + (size_t)b * XROW * NC + NC);
    v4f*       ob4    = (v4f*)      (out + (size_t)b * OROW * NC + NC);
    for (int e = t; e < 256 * NC4; e += 1024) {
        const int r = e / NC4;
        const int c = e - r * NC4;
        ob4[(size_t)r * NC4 + c] = nodes4[(size_t)s_idx[768 + r] * NC4 + c];
    }
}

// ---------------------------------------------------------------------------
extern "C" void kernel_launch(void* const* d_in, const int* in_sizes, int n_in,
                              void* d_out, int out_size, void* d_ws, size_t ws_size,
                              hipStream_t stream) {
    const float* x    = (const float*)d_in[0];   // (64,1025,384)
    const float* cls  = (const float*)d_in[1];   // (64,1,384)
    const float* eagg = (const float*)d_in[2];   // (64,1024,384)
    const float* W    = (const float*)d_in[3];   // (384,384)
    // d_in[4] = pool (==1 in this problem); pool=1 path implemented.
    float* out = (float*)d_out;                  // (64,257,384)

    // workspace layout (floats), ~2.4 MB total
    float* ws      = (float*)d_ws;
    float* ea_part = ws;                               // 64*8*384
    float* na_part = ea_part + NB * NPART * NC;        // 64*8*384
    float* ea      = na_part + NB * NPART * NC;        // 64*384
    float* ci      = ea + NB * NC;                     // 64*384
    float* ca      = ci + NB * NC;                     // 64*1024
    float* ni      = ca + NB * NHW;                    // 64*1024
    float* wts     = ni + NB * NHW;                    // 64*384

    k_ea_partial <<<dim3(NB, NPART), NC4, 0, stream>>>(eagg, ea_part);
    k_ea_combine <<<NB, NC, 0, stream>>>(ea_part, ea);
    k_ci_softmax <<<1, NC, 0, stream>>>(ea, ci);
    k_channel_agg<<<dim3(NB, NHW / 8), 256, 0, stream>>>(x, ci, ca);
    k_ni_softmax <<<NHW / 256, 256, 0, stream>>>(ca, ni);
    k_na_partial <<<dim3(NB, NPART), NC4, 0, stream>>>(x, ni, na_part);
    k_na_combine <<<NB, NC, 0, stream>>>(na_part, wts);
    k_cls_wmma   <<<dim3(NB / 16, NC / 16), 32, 0, stream>>>(cls, W, wts, out);
    k_sort_pool  <<<NB, NHW, 0, stream>>>(x, ca, out);
}